// MechanicsFunctionsMultiBlock_59167469469779
// MI455X (gfx1250) — compile-verified
//
#include <hip/hip_runtime.h>
#include <stdint.h>

#define BLK 256
typedef float v4f __attribute__((ext_vector_type(4)));

// Pass 1: invert the two block-index lists into a per-element material id (byte).
__global__ __launch_bounds__(BLK) void mat_scatter_kernel(
    const int* __restrict__ b0, int n0,
    const int* __restrict__ b1, int n1,
    unsigned char* __restrict__ mat)
{
    int i = blockIdx.x * BLK + threadIdx.x;
    if (i < n0)           mat[b0[i]] = 0;
    else if (i < n0 + n1) mat[b1[i - n0]] = 1;
}

// Pass 2: one lane per element, elements in order -> fully streamed I/O.
__global__ __launch_bounds__(BLK) void hess_kernel(
    const float* __restrict__ U,      // [N,2]
    const float* __restrict__ state,  // [E,1,1]
    const int*   __restrict__ conns,  // [E,3]
    const float* __restrict__ sgrad,  // [E,1,3,2]
    const float* __restrict__ vols,   // [E,1]
    const unsigned char* __restrict__ mat, // [E]
    float*       __restrict__ out,    // [E,3,2,3,2]
    int E)
{
    __shared__ float gt[BLK * 6];                 // async-staged shapeGrads
    __shared__ float hs[(BLK / 32) * 32 * 37];    // store-transpose, stride 37 (conflict-free)

    const int tid = threadIdx.x;
    const int e   = blockIdx.x * BLK + tid;
    const int es  = (e < E) ? e : (E - 1);        // clamped: all lanes load in-bounds

    // ---- CDNA5 async global->LDS staging of this lane's shapeGrads (24B) ----
    {
        uint64_t ga  = (uint64_t)(uintptr_t)(sgrad + (size_t)es * 6);
        uint32_t lds = (uint32_t)(uintptr_t)(&gt[tid * 6]);   // low 32 bits = LDS byte offset
        asm volatile(
            "global_load_async_to_lds_b64 %0, %1, off\n\t"
            "global_load_async_to_lds_b64 %0, %1, off offset:8\n\t"
            "global_load_async_to_lds_b64 %0, %1, off offset:16"
            :: "v"(lds), "v"(ga) : "memory");
    }

    // Prefetch streaming arrays 8 tiles ahead (global_prefetch_b8).
    {
        int ep = e + BLK * 8;
        if (ep < E) {
            __builtin_prefetch(sgrad + (size_t)ep * 6, 0, 0);
            __builtin_prefetch(conns + (size_t)ep * 3, 0, 0);
        }
    }

    // Streaming loads + the (L2-resident) random U gather, overlapped with the
    // in-flight async LDS transfer.
    const int   c0 = conns[(size_t)es * 3 + 0];
    const int   c1 = conns[(size_t)es * 3 + 1];
    const int   c2 = conns[(size_t)es * 3 + 2];
    const float q   = state[es];
    const float vol = vols[es];
    const int   mb  = mat[es];
    const float2 u0 = ((const float2*)U)[c0];
    const float2 u1 = ((const float2*)U)[c1];
    const float2 u2 = ((const float2*)U)[c2];

    const float lam    = mb ? 2.0f : 1.0f;
    const float mu     = mb ? 1.0f : 0.5f;
    const float mu_eff = mu * (1.0f + 0.01f * q);

    // Consume the async transfer (per-wave counter; lanes read only own slots).
    asm volatile("s_wait_asynccnt 0" ::: "memory");
    const float G00 = gt[tid * 6 + 0], G01 = gt[tid * 6 + 1];
    const float G10 = gt[tid * 6 + 2], G11 = gt[tid * 6 + 3];
    const float G20 = gt[tid * 6 + 4], G21 = gt[tid * 6 + 5];

    // gradU_{cd} = sum_n u_{n,c} G_{n,d};  F = I + gradU
    const float a00 = u0.x * G00 + u1.x * G10 + u2.x * G20;
    const float a01 = u0.x * G01 + u1.x * G11 + u2.x * G21;
    const float a10 = u0.y * G00 + u1.y * G10 + u2.y * G20;
    const float a11 = u0.y * G01 + u1.y * G11 + u2.y * G21;
    const float F00 = 1.0f + a00, F01 = a01, F10 = a10, F11 = 1.0f + a11;

    const float J    = F00 * F11 - F01 * F10;
    const float lj   = logf(J);
    const float ct   = lam * lj - mu_eff;            // c~ = lam*logJ - mu_eff
    const float invJ = 1.0f / J;
    const float alpha = (lam - ct) * invJ * invJ;    // (mu_eff + lam*(1-logJ))/J^2
    const float beta  = ct * invJ;

    // b_{n,c} = G_{n,:} . Cof(F)_{c,:}   (Cof = [[F11,-F10],[-F01,F00]])
    const float bx[3] = { G00 * F11 - G01 * F10,
                          G10 * F11 - G11 * F10,
                          G20 * F11 - G21 * F10 };
    const float by[3] = { G01 * F00 - G00 * F01,
                          G11 * F00 - G10 * F01,
                          G21 * F00 - G20 * F01 };
    const float Gx[3] = { G00, G10, G20 };
    const float Gy[3] = { G01, G11, G21 };

    const float vm = vol * mu_eff, va = vol * alpha, vb = vol * beta;

    // H[n,c,m,e] = vol*( mu_eff*d_ce*(Gn.Gm) + alpha*b_nc*b_me + beta*eps_ce*T_nm )
    float h[36];
#pragma unroll
    for (int n = 0; n < 3; ++n) {
#pragma unroll
        for (int m = 0; m < 3; ++m) {
            const float S = Gx[n] * Gx[m] + Gy[n] * Gy[m];
            const float T = Gx[n] * Gy[m] - Gy[n] * Gx[m];
            const int base = n * 12 + m * 2;
            h[base + 0]     = vm * S + va * bx[n] * bx[m];
            h[base + 1]     = va * bx[n] * by[m] + vb * T;
            h[base + 6 + 0] = va * by[n] * bx[m] - vb * T;
            h[base + 6 + 1] = vm * S + va * by[n] * by[m];
        }
    }

    // ---- wave-local LDS transpose -> dense 512B-per-instruction NT stores ----
    const int wave = tid >> 5, lane = tid & 31;
    float* hw = hs + wave * (32 * 37);
#pragma unroll
    for (int k = 0; k < 36; ++k)
        hw[lane * 37 + k] = h[k];                 // stride 37: bank-conflict-free

    __builtin_amdgcn_wave_barrier();              // cross-lane LDS dep: pin scheduling
    asm volatile("" ::: "memory");                // ...and compiler memory ordering

    const int webase = blockIdx.x * BLK + wave * 32;   // first element of this wave
    v4f* o4 = (v4f*)(out + (size_t)webase * 36);
#pragma unroll
    for (int t = 0; t < 9; ++t) {
        const int f4   = lane + 32 * t;           // float4 index in wave's 4608B span
        const int elem = f4 / 9;                  // 36 = 4*9: whole float4 in one element
        const int kb   = 4 * (f4 % 9);
        if (webase + elem < E) {
            const float* src = hw + elem * 37 + kb;
            v4f v = { src[0], src[1], src[2], src[3] };
            __builtin_nontemporal_store(v, o4 + f4);   // write-once: bypass L2 residency
        }
    }
}

extern "C" void kernel_launch(void* const* d_in, const int* in_sizes, int n_in,
                              void* d_out, int out_size, void* d_ws, size_t ws_size,
                              hipStream_t stream)
{
    // setup_inputs order: U, coords, state, conns, shapes, shapeGrads, vols, blocks0, blocks1
    const float* U     = (const float*)d_in[0];
    const float* state = (const float*)d_in[2];
    const int*   conns = (const int*)  d_in[3];
    const float* sgrad = (const float*)d_in[5];
    const float* vols  = (const float*)d_in[6];
    const int*   b0    = (const int*)  d_in[7];
    const int*   b1    = (const int*)  d_in[8];
    const int n0 = in_sizes[7];
    const int n1 = in_sizes[8];
    const int E  = in_sizes[3] / 3;

    unsigned char* mat = (unsigned char*)d_ws;   // E bytes of scratch

    const int total = n0 + n1;
    mat_scatter_kernel<<<(total + BLK - 1) / BLK, BLK, 0, stream>>>(b0, n0, b1, n1, mat);
    hess_kernel<<<(E + BLK - 1) / BLK, BLK, 0, stream>>>(
        U, state, conns, sgrad, vols, mat, (float*)d_out, E);
}